// HARP_12446815224372
// MI455X (gfx1250) — compile-verified
//
#include <hip/hip_runtime.h>

// ---------------- problem constants (match reference) ----------------
#define BATCH   8
#define NNODES  10000
#define NEDGES  320000
#define SLOPE   0.02f
#define TPB     256

typedef __attribute__((ext_vector_type(2))) float v2f;
typedef __attribute__((ext_vector_type(8))) float v8f;

static __device__ __forceinline__ float leaky(float v) {
    return v >= 0.0f ? v : SLOPE * v;
}

// ---------------- simple fill ----------------
__global__ void k_fill(float* __restrict__ p, float v, int n) {
    int i = blockIdx.x * blockDim.x + threadIdx.x;
    if (i < n) p[i] = v;
}

// ---------------- degree accumulation: deg pre-filled with 1.0 (self loop) ----------------
__global__ void k_degree(const int* __restrict__ ei, const float* __restrict__ cap,
                         float* __restrict__ deg) {
    int i = blockIdx.x * blockDim.x + threadIdx.x;
    if (i >= BATCH * NEDGES) return;
    int b = i / NEDGES;
    int e = i - b * NEDGES;
    int col = ei[NEDGES + e];
    atomicAdd(&deg[b * NNODES + col], cap[i]);
}

// ---------------- deg -> dis (rsqrt in place) and self-loop norm = dis^2 ----------------
__global__ void k_rsqrt(float* __restrict__ dis, float* __restrict__ nself) {
    int i = blockIdx.x * blockDim.x + threadIdx.x;
    if (i >= BATCH * NNODES) return;
    float d = dis[i];                 // >= 1.0 always (self loop weight folded in)
    float r = rsqrtf(d);
    dis[i]   = r;
    nself[i] = r * r;
}

// ---------------- per-edge norm = dis[row]*cap*dis[col] (computed once, reused 4x) ----------
__global__ void k_edgenorm(const int* __restrict__ ei, const float* __restrict__ cap,
                           const float* __restrict__ dis, float* __restrict__ nrm) {
    int i = blockIdx.x * blockDim.x + threadIdx.x;
    if (i >= BATCH * NEDGES) return;
    int b = i / NEDGES;
    int e = i - b * NEDGES;
    int r = ei[e];
    int c = ei[NEDGES + e];
    nrm[i] = dis[b * NNODES + r] * cap[i] * dis[b * NNODES + c];
}

// ---------------- pack node features [B,N,2] -> padded [B,N,4] ----------------
__global__ void k_pack(const float* __restrict__ nf, float* __restrict__ x) {
    int i = blockIdx.x * blockDim.x + threadIdx.x;
    if (i >= BATCH * NNODES) return;
    float4 v;
    v.x = nf[2 * i + 0];
    v.y = nf[2 * i + 1];
    v.z = 0.0f;
    v.w = 0.0f;
    reinterpret_cast<float4*>(x)[i] = v;
}

// ---------------- WMMA node transform: h[16 nodes] = A(16x4) * Wpad(4x16) -------------------
// A layout (32-bit 16x4): lane<16 holds (K=0,K=1), lane>=16 holds (K=2,K=3) for M = lane&15.
// B layout (4x16):        lane<16 holds (K=0,K=1), lane>=16 holds (K=2,K=3) for N = lane&15.
// D layout (16x16 f32, 8 VGPRs): VGPR j -> M = j + 8*(lane>=16), N = lane&15.
// DIN/DOUT are compile-time: W operand built with clamped unconditional loads + cndmask
// selects, so there is no EXEC-divergent control flow feeding the WMMA.
template <int DIN, int DOUT>
__global__ void k_xform_wmma(const float* __restrict__ x, const float* __restrict__ W,
                             float* __restrict__ h) {
    int gtid = blockIdx.x * blockDim.x + threadIdx.x;
    int wave = gtid >> 5;
    int lane = gtid & 31;
    const int tilesPerBatch = NNODES / 16;          // 625, exact
    int b = wave / tilesPerBatch;
    int t = wave - b * tilesPerBatch;               // grid sized exactly: 8*625 waves
    int base = (b * NNODES + t * 16) * 4;           // padded stride-4 float slab
    int half = lane >> 4;                           // 0: K=0,1   1: K=2,3
    int m    = lane & 15;                           // row (A) / col (B)

    const float* ap = x + base + m * 4 + half * 2;
    v2f a;
    a.x = ap[0];
    a.y = ap[1];

    // B operand: W padded to 4x16 with zeros, branch-free
    const int maxi = DIN * DOUT - 1;
    int k0 = half * 2;
    int i0 = (k0 + 0) * DOUT + m;
    int i1 = (k0 + 1) * DOUT + m;
    float w0 = W[i0 > maxi ? maxi : i0];            // unconditional clamped load
    float w1 = W[i1 > maxi ? maxi : i1];
    v2f bb;
    bb.x = (m < DOUT && (k0 + 0) < DIN) ? w0 : 0.0f;
    bb.y = (m < DOUT && (k0 + 1) < DIN) ? w1 : 0.0f;

    v8f c = {};
    v8f d = __builtin_amdgcn_wmma_f32_16x16x4_f32(false, a, false, bb,
                                                  (short)0, c, false, false);

    if (m < 4) {                                    // only first 4 columns of D are real
#pragma unroll
        for (int j = 0; j < 8; ++j) {
            int row = half * 8 + j;
            h[base + row * 4 + m] = d[j];
        }
    }
}

// ---------------- aggregation init with the self-loop term (no atomics needed) ---------------
__global__ void k_agg_init(const float* __restrict__ nself, const float* __restrict__ h,
                           float* __restrict__ agg) {
    int i = blockIdx.x * blockDim.x + threadIdx.x;
    if (i >= BATCH * NNODES) return;
    float s = nself[i];
    float4 hv = reinterpret_cast<const float4*>(h)[i];
    float4 o;
    o.x = s * hv.x; o.y = s * hv.y; o.z = s * hv.z; o.w = s * hv.w;
    reinterpret_cast<float4*>(agg)[i] = o;
}

// ---------------- edge scatter: agg[col] += norm * h[row] ----------------
__global__ void k_scatter(const int* __restrict__ ei, const float* __restrict__ nrm,
                          const float* __restrict__ h, float* __restrict__ agg) {
    int i = blockIdx.x * blockDim.x + threadIdx.x;
    if (i >= BATCH * NEDGES) return;
    int b = i / NEDGES;
    int e = i - b * NEDGES;
    int r = ei[e];
    int c = ei[NEDGES + e];
    float w = nrm[i];
    float4 hv = reinterpret_cast<const float4*>(h)[b * NNODES + r];  // one 16B gather
    float* ap = agg + (b * NNODES + c) * 4;
    atomicAdd(ap + 0, w * hv.x);
    atomicAdd(ap + 1, w * hv.y);
    atomicAdd(ap + 2, w * hv.z);
    atomicAdd(ap + 3, w * hv.w);
}

// ---------------- bias + leaky ReLU; write next-layer x (padded) and emb slab -----------------
__global__ void k_activate(const float* __restrict__ agg, const float* __restrict__ bias,
                           float* __restrict__ x, float* __restrict__ emb,
                           int dout, int off) {
    int i = blockIdx.x * blockDim.x + threadIdx.x;
    if (i >= BATCH * NNODES) return;
    float4 v = reinterpret_cast<const float4*>(agg)[i];
    float in[4] = { v.x, v.y, v.z, v.w };
    float out[4];
#pragma unroll
    for (int c = 0; c < 4; ++c) {
        float t = leaky(in[c] + (c < dout ? bias[c] : 0.0f));
        out[c] = (c < dout) ? t : 0.0f;
        if (c < dout) emb[i * 16 + off + c] = t;
    }
    float4 xo; xo.x = out[0]; xo.y = out[1]; xo.z = out[2]; xo.w = out[3];
    reinterpret_cast<float4*>(x)[i] = xo;
}

// ---------------- output: 4 threads per edge, one float4 each -------------------------------
__global__ void k_output(const int* __restrict__ ei, const float* __restrict__ cap,
                         const float* __restrict__ emb, float* __restrict__ out) {
    int i = blockIdx.x * blockDim.x + threadIdx.x;
    if (i >= BATCH * NEDGES * 4) return;
    int q = i & 3;
    int t = i >> 2;                  // b*NEDGES + e
    int b = t / NEDGES;
    int e = t - b * NEDGES;
    int r = ei[e];
    int c = ei[NEDGES + e];
    float4 av = reinterpret_cast<const float4*>(emb + (b * NNODES + r) * 16)[q];
    float4 bv = reinterpret_cast<const float4*>(emb + (b * NNODES + c) * 16)[q];
    float4 o;
    o.x = av.x + bv.x; o.y = av.y + bv.y; o.z = av.z + bv.z; o.w = av.w + bv.w;
    if (q == 3) o.w = cap[t];        // emb slot 15 is zero on both sides
    reinterpret_cast<float4*>(out)[i] = o;
}

// ---------------- host side ----------------
static inline int nblk(long long n) { return (int)((n + TPB - 1) / TPB); }

extern "C" void kernel_launch(void* const* d_in, const int* in_sizes, int n_in,
                              void* d_out, int out_size, void* d_ws, size_t ws_size,
                              hipStream_t stream) {
    const float* nf  = (const float*)d_in[0];   // [8,10000,2]
    const int*   ei  = (const int*)  d_in[1];   // [2,320000]
    const float* cap = (const float*)d_in[2];   // [8,320000]
    const float* Wl[4] = { (const float*)d_in[3], (const float*)d_in[5],
                           (const float*)d_in[7], (const float*)d_in[9] };
    const float* bl[4] = { (const float*)d_in[4], (const float*)d_in[6],
                           (const float*)d_in[8], (const float*)d_in[10] };
    float* out = (float*)d_out;

    const int dout_[4] = { 3, 4, 4, 4 };
    const int off_[4]  = { 0, 3, 7, 11 };

    const long long BN = (long long)BATCH * NNODES;      //    80 000
    const long long BE = (long long)BATCH * NEDGES;      // 2 560 000

    // workspace layout (floats), all 16B-aligned offsets
    float* ws    = (float*)d_ws;
    float* dis   = ws;                       // BN       (deg, then rsqrt in place)
    float* nself = dis   + BN;               // BN
    float* nrm   = nself + BN;               // BE
    float* x     = nrm   + BE;               // BN*4
    float* h     = x     + BN * 4;           // BN*4
    float* agg   = h     + BN * 4;           // BN*4
    float* emb   = agg   + BN * 4;           // BN*16
    // total = 4 960 000 floats = 19.84 MB

    // 1) init
    k_fill<<<nblk(BN), TPB, 0, stream>>>(dis, 1.0f, (int)BN);          // self-loop weight
    k_fill<<<nblk(BN * 16), TPB, 0, stream>>>(emb, 0.0f, (int)(BN * 16));

    // 2) degree -> dis, self-loop norm
    k_degree<<<nblk(BE), TPB, 0, stream>>>(ei, cap, dis);
    k_rsqrt<<<nblk(BN), TPB, 0, stream>>>(dis, nself);

    // 3) per-edge norm (reused by all layers)
    k_edgenorm<<<nblk(BE), TPB, 0, stream>>>(ei, cap, dis, nrm);

    // 4) pack features to padded float4 slab
    k_pack<<<nblk(BN), TPB, 0, stream>>>(nf, x);

    // 5) layers
    const int xformBlocks = BATCH * (NNODES / 16) * 32 / TPB;   // 5000 waves, exact grid
    for (int l = 0; l < 4; ++l) {
        switch (l) {
            case 0: k_xform_wmma<2, 3><<<xformBlocks, TPB, 0, stream>>>(x, Wl[0], h); break;
            case 1: k_xform_wmma<3, 4><<<xformBlocks, TPB, 0, stream>>>(x, Wl[1], h); break;
            case 2: k_xform_wmma<4, 4><<<xformBlocks, TPB, 0, stream>>>(x, Wl[2], h); break;
            default:k_xform_wmma<4, 4><<<xformBlocks, TPB, 0, stream>>>(x, Wl[3], h); break;
        }
        k_agg_init<<<nblk(BN), TPB, 0, stream>>>(nself, h, agg);
        k_scatter<<<nblk(BE), TPB, 0, stream>>>(ei, nrm, h, agg);
        k_activate<<<nblk(BN), TPB, 0, stream>>>(agg, bl[l], x, emb, dout_[l], off_[l]);
    }

    // 6) edge output
    k_output<<<nblk(BE * 4), TPB, 0, stream>>>(ei, cap, emb, out);
}